// SplineInter_91233695302105
// MI455X (gfx1250) — compile-verified
//
#include <hip/hip_runtime.h>

// 2D cubic B-spline interpolation, m = 1024x1024, PAD=2, omega=[0,1]^2.
// Gather-bound kernel: coeff table (4.2 MB) lives in L2/WGP$; x/out are
// streamed with non-temporal hints so they don't pollute the hot table.

#define ROW_STRIDE 1028              // M1 + 2*PAD
#define NELEM      (1028 * 1028)     // padded grid elements
#define GRID_M     1024.0f

// 4-float vector with only 4-byte alignment guarantee: lets the backend emit
// a single global_load_b128 per stencil row (dword-aligned b128 is legal on
// gfx1250) while remaining correct if it must split.
typedef float f4u  __attribute__((ext_vector_type(4), aligned(4)));
typedef float f4a  __attribute__((ext_vector_type(4)));   // 16B aligned
typedef float f2a  __attribute__((ext_vector_type(2)));   // 8B aligned

// Cubic B-spline pieces, same algebra as reference _b0(3-j, t-j), j=-1..2
__device__ __forceinline__ void bspline_weights(float t, float w[4]) {
    // j = -1 : xi = t+1 : (2 - xi)^3 = (1-t)^3
    float a = 2.0f - (t + 1.0f);
    w[0] = a * a * a;
    // j = 0  : xi = t   : (3t - 6)*t^2 + 4
    w[1] = fmaf(fmaf(3.0f, t, -6.0f), t * t, 4.0f);
    // j = 1  : xi = t-1 : -(3xi + 6)*xi^2 + 4
    float b = t - 1.0f;
    w[2] = fmaf(-fmaf(3.0f, b, 6.0f), b * b, 4.0f);
    // j = 2  : xi = t-2 : (2 + xi)^3 = t^3
    float c = 2.0f + (t - 2.0f);
    w[3] = c * c * c;
}

__device__ __forceinline__ float spline_eval(float x0, float x1,
                                             const float* __restrict__ coeffs) {
    // xn = x/h - 0.5 ; h = 2^-10 so x*1024 is exact, fma == mul+add here
    float xn0 = fmaf(x0, GRID_M, -0.5f);
    float xn1 = fmaf(x1, GRID_M, -0.5f);

    bool valid = (xn0 > -2.0f) & (xn0 < GRID_M) & (xn1 > -2.0f) & (xn1 < GRID_M);

    float f0 = floorf(xn0), f1 = floorf(xn1);
    int   P0 = (int)f0,     P1 = (int)f1;
    float t0 = xn0 - f0,    t1 = xn1 - f1;

    float w0[4], w1[4];
    bspline_weights(t0, w0);
    bspline_weights(t1, w1);

    int p = ROW_STRIDE * (2 + P0) + (2 + P1);   // flat base index

    // Gather the 4x4 stencil as four dword-aligned b128 row loads.
    // Row start clamped in-bounds (matches reference clip for all inputs that
    // actually reach the gather: points in [0,1]^2 have every index in range).
    f4u rows[4];
#pragma unroll
    for (int j0 = 0; j0 < 4; ++j0) {
        int rs = p + (j0 - 1) * ROW_STRIDE - 1;
        rs = max(0, min(rs, NELEM - 4));
        rows[j0] = *(const f4u*)(coeffs + rs);
    }

    // out = w0^T * V * w1
    float acc = 0.0f;
#pragma unroll
    for (int j0 = 0; j0 < 4; ++j0) {
        float s = rows[j0][0] * w1[0];
        s = fmaf(rows[j0][1], w1[1], s);
        s = fmaf(rows[j0][2], w1[2], s);
        s = fmaf(rows[j0][3], w1[3], s);
        acc = fmaf(s, w0[j0], acc);
    }
    return valid ? acc : 0.0f;
}

// 2 points per thread: one 16B coalesced NT load of x, 8 b128 gathers in
// flight, one 8B coalesced NT store. wave32: 256-thread block = 8 waves.
__global__ void __launch_bounds__(256)
spline_pair_kernel(const float* __restrict__ x, const float* __restrict__ coeffs,
                   float* __restrict__ out, int npairs) {
    int tid = blockIdx.x * blockDim.x + threadIdx.x;
    if (tid >= npairs) return;

    // streamed exactly once -> non-temporal so coeff table stays resident
    f4a q = __builtin_nontemporal_load((const f4a*)x + tid);

    float r0 = spline_eval(q[0], q[1], coeffs);
    float r1 = spline_eval(q[2], q[3], coeffs);

    f2a o; o[0] = r0; o[1] = r1;
    __builtin_nontemporal_store(o, (f2a*)out + tid);
}

// Remainder (N odd) — not hit for N = 2^21 but kept for generality.
__global__ void spline_tail_kernel(const float* __restrict__ x,
                                   const float* __restrict__ coeffs,
                                   float* __restrict__ out, int n) {
    if (threadIdx.x == 0 && blockIdx.x == 0) {
        int i = n - 1;
        out[i] = spline_eval(x[2 * i], x[2 * i + 1], coeffs);
    }
}

extern "C" void kernel_launch(void* const* d_in, const int* in_sizes, int n_in,
                              void* d_out, int out_size, void* d_ws, size_t ws_size,
                              hipStream_t stream) {
    const float* x      = (const float*)d_in[0];   // [N,2] float32
    const float* coeffs = (const float*)d_in[1];   // [1028,1028] float32
    float* out = (float*)d_out;                    // [N] float32

    int N = in_sizes[0] / 2;
    int npairs = N / 2;

    if (npairs > 0) {
        dim3 block(256);
        dim3 grid((npairs + 255) / 256);
        spline_pair_kernel<<<grid, block, 0, stream>>>(x, coeffs, out, npairs);
    }
    if (N & 1) {
        spline_tail_kernel<<<1, 32, 0, stream>>>(x, coeffs, out, N);
    }
}